// Dynamic_Prototype_16131897164203
// MI455X (gfx1250) — compile-verified
//
#include <hip/hip_runtime.h>
#include <math.h>

typedef __attribute__((ext_vector_type(2))) float v2f;
typedef __attribute__((ext_vector_type(8))) float v8f;

// Problem constants (reference: B=16, C=512, H=W=64 -> hw=4096, P=16)
#define NB 16
#define NC 512
#define NHW 4096
#define NP 16
#define SKP 516   // padded LDS row stride (516 mod 64 = 4 -> bank-conflict-free)

// ---------------------------------------------------------------------------
// K0: zero loss accumulators
// ---------------------------------------------------------------------------
__global__ void k_init(float* __restrict__ accs) {
  if (threadIdx.x < 8) accs[threadIdx.x] = 0.0f;
}

// ---------------------------------------------------------------------------
// K1: logits[b,n,p] = sum_d key[b,d,n] * W[p,d]
// one wave per 16-row n-tile, WMMA f32 16x16x4, K=512 (128 wmma ops)
// ---------------------------------------------------------------------------
__global__ __launch_bounds__(256) void k_logits(const float* __restrict__ key,
                                                const float* __restrict__ Wm,
                                                float* __restrict__ logits) {
  __shared__ float sW[NP * SKP];
  for (int i = threadIdx.x; i < NP * NC; i += 256) {
    int p = i >> 9, d = i & 511;
    sW[p * SKP + d] = Wm[i];
  }
  __syncthreads();

  const int wave = threadIdx.x >> 5;
  const int lane = threadIdx.x & 31;
  const int tile = blockIdx.x * 8 + wave;      // 4096 tiles total
  const int b = tile >> 8;                     // 256 tiles per batch
  const int n0 = (tile & 255) << 4;
  const int m  = lane & 15;                    // A: M index / B: col index
  const int kb = (lane >> 4) << 1;             // A/B: K sub-index base {0,2}
  const int hi = (lane >> 4) << 3;             // D: row offset {0,8}

  v8f acc = {};
  const float* kbase = key + (size_t)b * NC * NHW + n0 + m;
  for (int d0 = 0; d0 < NC; d0 += 4) {
    v2f a, bm;
    const float* kp = kbase + (size_t)(d0 + kb) * NHW;
    a.x = kp[0];
    a.y = kp[NHW];
    bm.x = sW[m * SKP + d0 + kb];
    bm.y = sW[m * SKP + d0 + kb + 1];
    acc = __builtin_amdgcn_wmma_f32_16x16x4_f32(false, a, false, bm,
                                                (short)0, acc, false, false);
  }
#pragma unroll
  for (int r = 0; r < 8; ++r) {
    int n = n0 + r + hi;
    logits[((size_t)b * NHW + n) * NP + m] = acc[r];
  }
}

// ---------------------------------------------------------------------------
// K2: per-(b,p) max and 1/sum(exp) over n (softmax over spatial axis)
// ---------------------------------------------------------------------------
__global__ __launch_bounds__(256) void k_stats(const float* __restrict__ logits,
                                               float* __restrict__ stats) {
  __shared__ float red[256];
  const int bp = blockIdx.x;                   // b*16+p
  const int b = bp >> 4, p = bp & 15;
  const float* col = logits + (size_t)b * NHW * NP + p;

  float mx = -1e30f;
  for (int n = threadIdx.x; n < NHW; n += 256) mx = fmaxf(mx, col[(size_t)n * NP]);
  red[threadIdx.x] = mx;
  __syncthreads();
  for (int s = 128; s > 0; s >>= 1) {
    if (threadIdx.x < s) red[threadIdx.x] = fmaxf(red[threadIdx.x], red[threadIdx.x + s]);
    __syncthreads();
  }
  mx = red[0];
  __syncthreads();

  float sum = 0.0f;
  for (int n = threadIdx.x; n < NHW; n += 256) sum += __expf(col[(size_t)n * NP] - mx);
  red[threadIdx.x] = sum;
  __syncthreads();
  for (int s = 128; s > 0; s >>= 1) {
    if (threadIdx.x < s) red[threadIdx.x] += red[threadIdx.x + s];
    __syncthreads();
  }
  if (threadIdx.x == 0) {
    stats[bp * 2]     = mx;
    stats[bp * 2 + 1] = 1.0f / red[0];
  }
}

// ---------------------------------------------------------------------------
// K2b: logits -> softmax weights in place
// ---------------------------------------------------------------------------
__global__ __launch_bounds__(256) void k_wnorm(float* __restrict__ logits,
                                               const float* __restrict__ stats) {
  size_t i = (size_t)blockIdx.x * 256 + threadIdx.x;  // over NB*NHW*NP = 1M
  if (i >= (size_t)NB * NHW * NP) return;
  int p = (int)(i & 15);
  int b = (int)(i >> 16);                             // NHW*NP = 65536
  float mx  = stats[(b * NP + p) * 2];
  float inv = stats[(b * NP + p) * 2 + 1];
  logits[i] = __expf(logits[i] - mx) * inv;
}

// ---------------------------------------------------------------------------
// K3: protos[b,p,d] = sum_n wgt[b,n,p] * key[b,d,n]   (one wave per (b,d))
// ---------------------------------------------------------------------------
__global__ __launch_bounds__(256) void k_protos(const float* __restrict__ key,
                                                const float* __restrict__ wgt,
                                                float* __restrict__ protos) {
  const int wave = threadIdx.x >> 5, lane = threadIdx.x & 31;
  const int id = blockIdx.x * 8 + wave;        // NB*NC = 8192 waves
  const int b = id >> 9, d = id & 511;
  const float* kp = key + ((size_t)b * NC + d) * NHW;
  const float* wp = wgt + (size_t)b * NHW * NP;

  float acc[NP];
#pragma unroll
  for (int p = 0; p < NP; ++p) acc[p] = 0.0f;

  for (int n = lane; n < NHW; n += 32) {
    float kv = kp[n];
    const float* w = wp + (size_t)n * NP;
#pragma unroll
    for (int p = 0; p < NP; ++p) acc[p] += kv * w[p];
  }
#pragma unroll
  for (int p = 0; p < NP; ++p) {
#pragma unroll
    for (int off = 16; off > 0; off >>= 1) acc[p] += __shfl_down(acc[p], off, 32);
  }
  if (lane == 0) {
    for (int p = 0; p < NP; ++p) protos[((size_t)b * NP + p) * NC + d] = acc[p];
  }
}

// ---------------------------------------------------------------------------
// K4: kn = l2norm(protos)  (one block per (b,p) vector of 512)
// ---------------------------------------------------------------------------
__global__ __launch_bounds__(256) void k_norm(const float* __restrict__ protos,
                                              float* __restrict__ kn) {
  __shared__ float red[256];
  const int bp = blockIdx.x;
  const float* v = protos + (size_t)bp * NC;
  float ss = 0.0f;
  for (int d = threadIdx.x; d < NC; d += 256) { float x = v[d]; ss += x * x; }
  red[threadIdx.x] = ss;
  __syncthreads();
  for (int s = 128; s > 0; s >>= 1) {
    if (threadIdx.x < s) red[threadIdx.x] += red[threadIdx.x + s];
    __syncthreads();
  }
  float inv = 1.0f / fmaxf(sqrtf(red[0]), 1e-12f);
  for (int d = threadIdx.x; d < NC; d += 256) kn[(size_t)bp * NC + d] = v[d] * inv;
}

// ---------------------------------------------------------------------------
// K5: diversity + compactness losses (raw sums into accs[2] / accs[1])
// ---------------------------------------------------------------------------
__global__ __launch_bounds__(256) void k_losses(const float* __restrict__ kn,
                                                float* __restrict__ accs) {
  __shared__ float sK[NP * SKP];
  __shared__ float red[256];
  const int b = blockIdx.x;
  for (int i = threadIdx.x; i < NP * NC; i += 256) {
    int p = i >> 9, d = i & 511;
    sK[p * SKP + d] = kn[(size_t)b * NP * NC + i];
  }
  __syncthreads();

  // diversity: 120 strict-upper-triangle pairs
  float dsum = 0.0f;
  for (int pr = threadIdx.x; pr < 120; pr += 256) {
    int i = 0, j = 0, c = pr;
    for (i = 0; i < NP; ++i) {
      int cnt = NP - 1 - i;
      if (c < cnt) { j = i + 1 + c; break; }
      c -= cnt;
    }
    float s = 0.0f;
    for (int d = 0; d < NC; ++d) {
      float df = sK[i * SKP + d] - sK[j * SKP + d];
      s += df * df;
    }
    float dist = 1.0f - s;
    dsum += dist > 0.0f ? dist : 0.0f;
  }
  red[threadIdx.x] = dsum;
  __syncthreads();
  for (int s = 128; s > 0; s >>= 1) {
    if (threadIdx.x < s) red[threadIdx.x] += red[threadIdx.x + s];
    __syncthreads();
  }
  if (threadIdx.x == 0) atomicAdd(&accs[2], red[0]);

  // compactness: sum over (p,d) of (kn[b]-kn[b-1])^2, for b>=1
  if (b >= 1) {
    __syncthreads();
    float cs = 0.0f;
    const float* prev = kn + (size_t)(b - 1) * NP * NC;
    for (int i = threadIdx.x; i < NP * NC; i += 256) {
      int p = i >> 9, d = i & 511;
      float df = sK[p * SKP + d] - prev[i];
      cs += df * df;
    }
    red[threadIdx.x] = cs;
    __syncthreads();
    for (int s = 128; s > 0; s >>= 1) {
      if (threadIdx.x < s) red[threadIdx.x] += red[threadIdx.x + s];
      __syncthreads();
    }
    if (threadIdx.x == 0) atomicAdd(&accs[1], red[0]);
  }
}

// ---------------------------------------------------------------------------
// K6: score = q_tok @ kn^T (WMMA), softmax over P (shfl width 16), argmax
// ---------------------------------------------------------------------------
__global__ __launch_bounds__(256) void k_score(const float* __restrict__ query,
                                               const float* __restrict__ kn,
                                               float* __restrict__ wbuf,
                                               int* __restrict__ top1) {
  __shared__ float sK[NP * SKP];
  const int b = blockIdx.x >> 5;
  for (int i = threadIdx.x; i < NP * NC; i += 256) {
    int p = i >> 9, d = i & 511;
    sK[p * SKP + d] = kn[(size_t)b * NP * NC + i];
  }
  __syncthreads();

  const int wave = threadIdx.x >> 5, lane = threadIdx.x & 31;
  const int n0 = (((blockIdx.x & 31) << 3) + wave) << 4;
  const int m = lane & 15, kb = (lane >> 4) << 1, hi = (lane >> 4) << 3;

  v8f acc = {};
  const float* qb = query + (size_t)b * NC * NHW + n0 + m;
  for (int d0 = 0; d0 < NC; d0 += 4) {
    v2f a, bm;
    const float* qp = qb + (size_t)(d0 + kb) * NHW;
    a.x = qp[0];
    a.y = qp[NHW];
    bm.x = sK[m * SKP + d0 + kb];
    bm.y = sK[m * SKP + d0 + kb + 1];
    acc = __builtin_amdgcn_wmma_f32_16x16x4_f32(false, a, false, bm,
                                                (short)0, acc, false, false);
  }

#pragma unroll
  for (int r = 0; r < 8; ++r) {
    float v = acc[r];
    float mx = v;
    int idx = m;
#pragma unroll
    for (int off = 8; off > 0; off >>= 1) {
      float ov = __shfl_xor(mx, off, 16);
      int oi = __shfl_xor(idx, off, 16);
      if (ov > mx) { mx = ov; idx = oi; }
    }
    float e = __expf(v - mx);
    float s = e;
#pragma unroll
    for (int off = 8; off > 0; off >>= 1) s += __shfl_xor(s, off, 16);
    float w = e / s;
    int n = n0 + r + hi;
    wbuf[((size_t)b * NHW + n) * NP + m] = w;
    if (m == 0) top1[b * NHW + n] = idx;
  }
}

// ---------------------------------------------------------------------------
// K7: newQuery = w @ kn (WMMA, K=P), l2norm rows, +q_tok, transpose-store,
//     featureLoss accumulation
// ---------------------------------------------------------------------------
__global__ __launch_bounds__(256) void k_newq(const float* __restrict__ query,
                                              const float* __restrict__ kn,
                                              const float* __restrict__ wbuf,
                                              const int* __restrict__ top1,
                                              float* __restrict__ outUQ,
                                              float* __restrict__ accs) {
  __shared__ float sK[NP * SKP];
  __shared__ float tb[8][16][17];
  const int b = blockIdx.x >> 5;
  for (int i = threadIdx.x; i < NP * NC; i += 256) {
    int p = i >> 9, d = i & 511;
    sK[p * SKP + d] = kn[(size_t)b * NP * NC + i];
  }
  __syncthreads();

  const int wave = threadIdx.x >> 5, lane = threadIdx.x & 31;
  const int n0 = (((blockIdx.x & 31) << 3) + wave) << 4;
  const int m = lane & 15, kb = (lane >> 4) << 1, hi = (lane >> 4) << 3;

  // A matrix (softmax weights) in WMMA A-layout, K = p in 4 chunks
  v2f aw[4];
  const float* wrow = wbuf + ((size_t)b * NHW + n0 + m) * NP;
#pragma unroll
  for (int kc = 0; kc < 4; ++kc) {
    aw[kc].x = wrow[4 * kc + kb];
    aw[kc].y = wrow[4 * kc + kb + 1];
  }

  // pass 1: per-row sum of squares of newQuery
  float ssq[8];
#pragma unroll
  for (int r = 0; r < 8; ++r) ssq[r] = 0.0f;
  for (int dt = 0; dt < 32; ++dt) {
    const int d0 = dt * 16;
    v8f acc = {};
#pragma unroll
    for (int kc = 0; kc < 4; ++kc) {
      v2f bm;
      int p0 = 4 * kc + kb;
      bm.x = sK[p0 * SKP + d0 + m];
      bm.y = sK[(p0 + 1) * SKP + d0 + m];
      acc = __builtin_amdgcn_wmma_f32_16x16x4_f32(false, aw[kc], false, bm,
                                                  (short)0, acc, false, false);
    }
#pragma unroll
    for (int r = 0; r < 8; ++r) ssq[r] += acc[r] * acc[r];
  }
  float invn[8];
#pragma unroll
  for (int r = 0; r < 8; ++r) {
    float s = ssq[r];
#pragma unroll
    for (int off = 8; off > 0; off >>= 1) s += __shfl_xor(s, off, 16);
    invn[r] = 1.0f / fmaxf(sqrtf(s), 1e-12f);
  }

  const int t1 = top1[b * NHW + n0 + m];  // top-1 prototype for this lane's n
  float floss = 0.0f;

  // pass 2: recompute, normalize, transpose via LDS, add q_tok, store
  for (int dt = 0; dt < 32; ++dt) {
    const int d0 = dt * 16;
    v8f acc = {};
#pragma unroll
    for (int kc = 0; kc < 4; ++kc) {
      v2f bm;
      int p0 = 4 * kc + kb;
      bm.x = sK[p0 * SKP + d0 + m];
      bm.y = sK[(p0 + 1) * SKP + d0 + m];
      acc = __builtin_amdgcn_wmma_f32_16x16x4_f32(false, aw[kc], false, bm,
                                                  (short)0, acc, false, false);
    }
    __syncthreads();
#pragma unroll
    for (int r = 0; r < 8; ++r) tb[wave][r + hi][m] = acc[r] * invn[r];
    __syncthreads();
    // transposed: lane m -> n, r2+hi -> d   => coalesced along n
#pragma unroll
    for (int r2 = 0; r2 < 8; ++r2) {
      int dloc = r2 + hi;
      float v = tb[wave][m][dloc];
      size_t gidx = ((size_t)b * NC + d0 + dloc) * NHW + n0 + m;
      float q = query[gidx];
      outUQ[gidx] = v + q;
      float pos = sK[t1 * SKP + d0 + dloc];
      float df = q - pos;
      floss += df * df;
    }
  }
#pragma unroll
  for (int off = 16; off > 0; off >>= 1) floss += __shfl_xor(floss, off, 32);
  if (lane == 0) atomicAdd(&accs[0], floss);
}

// ---------------------------------------------------------------------------
// K8: finalize scalar losses
// ---------------------------------------------------------------------------
__global__ void k_final(const float* __restrict__ accs, float* __restrict__ outS) {
  if (threadIdx.x == 0) {
    outS[0] = accs[0] / ((float)NB * (float)NHW * (float)NC);       // featureLoss
    outS[1] = accs[1] / ((float)(NB - 1) * (float)NP);              // compactnessLoss
    outS[2] = accs[2] * (2.0f / ((float)NP * (float)(NP - 1))) / (float)NB; // diversityLoss
  }
}

// ---------------------------------------------------------------------------
extern "C" void kernel_launch(void* const* d_in, const int* in_sizes, int n_in,
                              void* d_out, int out_size, void* d_ws, size_t ws_size,
                              hipStream_t stream) {
  const float* key   = (const float*)d_in[0];
  const float* query = (const float*)d_in[1];
  const float* Wm    = (const float*)d_in[2];

  float* out        = (float*)d_out;
  float* protos_out = out + (size_t)NB * NC * NHW;             // 33,554,432
  float* outS       = protos_out + (size_t)NB * NP * NC;       // +131,072

  float* ws     = (float*)d_ws;
  float* logits = ws;                        // 1,048,576 f32 (reused as wbuf)
  float* stats  = ws + 1048576;              // 512 f32
  float* knbuf  = ws + 1049088;              // 131,072 f32
  int*   top1   = (int*)(ws + 1180160);      // 65,536 i32
  float* accs   = ws + 1245696;              // 8 f32

  k_init  <<<1,    32,  0, stream>>>(accs);
  k_logits<<<512,  256, 0, stream>>>(key, Wm, logits);
  k_stats <<<256,  256, 0, stream>>>(logits, stats);
  k_wnorm <<<4096, 256, 0, stream>>>(logits, stats);
  k_protos<<<1024, 256, 0, stream>>>(key, logits, protos_out);
  k_norm  <<<256,  256, 0, stream>>>(protos_out, knbuf);
  k_losses<<<16,   256, 0, stream>>>(knbuf, accs);
  k_score <<<512,  256, 0, stream>>>(query, knbuf, logits /*wbuf*/, top1);
  k_newq  <<<512,  256, 0, stream>>>(query, knbuf, logits /*wbuf*/, top1, out, accs);
  k_final <<<1,    32,  0, stream>>>(accs, outS);
}